// RGCN_node_19971597926925
// MI455X (gfx1250) — compile-verified
//
#include <hip/hip_runtime.h>
#include <math.h>

typedef __attribute__((ext_vector_type(2))) float v2f;
typedef __attribute__((ext_vector_type(8))) float v8f;
typedef __attribute__((ext_vector_type(4))) unsigned int u32x4;
typedef __attribute__((ext_vector_type(8))) int i32x8;
typedef __attribute__((ext_vector_type(4))) int i32x4;

#define NN   100000   // nodes
#define FDIM 64       // feature dim (= HID)
#define RREL 20       // relations
#define CCLS 2        // classes
#define WSTRIDE 80    // LDS row stride (64 data + 16 pad DWORDs -> bank-conflict-free)

#if defined(__has_builtin)
#  if __has_builtin(__builtin_amdgcn_tensor_load_to_lds) && \
      __has_builtin(__builtin_amdgcn_s_wait_tensorcnt)
#    define USE_TDM 1
#  endif
#endif
#ifndef USE_TDM
#  define USE_TDM 0
#endif

// ---------------------------------------------------------------- utilities
__global__ __launch_bounds__(256) void zero_kernel(float* __restrict__ p, long long n) {
  long long i = (long long)blockIdx.x * 256 + threadIdx.x;
  if (i < n) p[i] = 0.0f;
}

// cnt[r, dst] += 1 over all edges (same for both layers)
__global__ __launch_bounds__(256) void count_kernel(const int* __restrict__ dst,
                                                    const int* __restrict__ et,
                                                    float* __restrict__ cnt, int nE) {
  int e = blockIdx.x * 256 + threadIdx.x;
  if (e >= nE) return;
  atomicAdd(&cnt[(long long)et[e] * NN + dst[e]], 1.0f);
}

// S[r-rbase, dst, f] += x[src, f] / max(cnt[r,dst],1)   for edges with r in chunk
__global__ __launch_bounds__(256) void scatter_kernel(
    const float* __restrict__ x, const int* __restrict__ src,
    const int* __restrict__ dst, const int* __restrict__ et,
    const float* __restrict__ cnt, float* __restrict__ S,
    int nE, int rbase, int rcount) {
  long long tid = (long long)blockIdx.x * 256 + threadIdx.x;
  long long e = tid >> 6;
  int f = (int)(tid & 63);
  if (e >= nE) return;
  int r = et[e] - rbase;
  if (r < 0 || r >= rcount) return;
  int d = dst[e];
  float cv = cnt[(long long)(r + rbase) * NN + d];
  float norm = 1.0f / fmaxf(cv, 1.0f);
  float val = x[(long long)src[e] * FDIM + f] * norm;
  atomicAdd(&S[((long long)r * NN + d) * FDIM + f], val);
}

// ------------------------------------------------- weight staging into LDS
// Copies one 64x64 f32 weight matrix W[r] (16 KB, contiguous) into LDS laid
// out as WT[k*80 + n]: 64 data DWORDs + 16 pad DWORDs per row, produced by
// the TDM's pad feature (pad_interval=64 DW, pad_amount=16 DW). The stride-80
// layout puts the two half-waves' B-fragment reads on disjoint bank sets.
__device__ __forceinline__ void stage_weights(const float* __restrict__ Br,
                                              float* __restrict__ WT) {
#if USE_TDM
  if (threadIdx.x < 32) {  // one wave issues the DMA; EXEC is ignored by TDM
    unsigned ldsOff = (unsigned)(unsigned long long)
        (__attribute__((address_space(3))) void*)WT;
    unsigned long long ga = (unsigned long long)Br;
    // D# group0: count=1 | lds_addr | global_addr[56:0] | type=2
    u32x4 g0 = { 1u,
                 ldsOff,
                 (unsigned)ga,
                 (unsigned)(((ga >> 32) & 0x01FFFFFFull)) | (2u << 30) };
    // D# group1: wg_mask=0, data_size=2(4B), pad_enable=1,
    //            pad_interval=5(64 DW), pad_amount=15(16 DW),
    //            tensor_dim0=4096, tensor_dim1=1, tile_dim0=4096, tile_dim1=1,
    //            tensor_dim0_stride=4096
    i32x8 g1 = { (int)0x1F520000u,
                 (int)(4096u << 16),   // tensor_dim0 (bits 79:48)
                 (int)(1u << 16),      // tensor_dim1 (bits 111:80)
                 (int)(4096u << 16),   // tile_dim0   (bits 127:112)
                 1,                    // tile_dim1   (bits 143:128)
                 4096,                 // tensor_dim0_stride (bits 207:160)
                 0, 0 };
    i32x4 gz = { 0, 0, 0, 0 };
#if __clang_major__ >= 23
    i32x8 g4 = { 0, 0, 0, 0, 0, 0, 0, 0 };
    __builtin_amdgcn_tensor_load_to_lds(g0, g1, gz, gz, g4, 0);
#else
    __builtin_amdgcn_tensor_load_to_lds(g0, g1, gz, gz, 0);
#endif
    __builtin_amdgcn_s_wait_tensorcnt(0);
  }
#else
  for (int i = threadIdx.x; i < 4096; i += 256)
    WT[(i >> 6) * WSTRIDE + (i & 63)] = Br[i];
#endif
}

// ---------------------------------------------------------------- WMMA GEMM
// out[nrows,64] (+)= sum_{r<nrel} A[r][nrows,64] @ B[r][64,64]  (+ bias row)
// One wave per 16-row tile; 4 v8f accumulators cover all 64 output columns.
// B tiles staged in LDS (TDM DMA), A streamed from global; fp32 matrix pipe
// V_WMMA_F32_16X16X4_F32, K fully unrolled (16 steps x 4 WMMA).
__global__ __launch_bounds__(256) void gemm16_kernel(
    const float* __restrict__ A, long long abatch, int nrel,
    const float* __restrict__ B, const float* __restrict__ bias,
    float* __restrict__ out, int nrows, int accumulate) {
  __shared__ float WT[64 * WSTRIDE];  // 20 KB
  int wave = threadIdx.x >> 5;
  int lane = threadIdx.x & 31;
  int row0 = (blockIdx.x * 8 + wave) * 16;
  bool active = row0 < nrows;
  int hl  = lane >> 4;   // half-wave select (K split)
  int l15 = lane & 15;   // M / N index within half

  v8f c0 = {}, c1 = {}, c2 = {}, c3 = {};

  for (int r = 0; r < nrel; ++r) {
    const float* __restrict__ Br = B + (long long)r * FDIM * FDIM;
    if (r + 1 < nrel)
      __builtin_prefetch(B + (long long)(r + 1) * FDIM * FDIM, 0, 1);
    if (r) __syncthreads();          // all waves done reading previous W[r]
    stage_weights(Br, WT);
    __syncthreads();                 // staged data visible to all waves
    if (active) {
      const float* __restrict__ Ar =
          A + (long long)r * abatch + (long long)(row0 + l15) * FDIM;
#pragma unroll
      for (int kk = 0; kk < FDIM; kk += 4) {
        int ka = kk + (hl << 1);
        // A fragment 16x4: lanes 0-15 hold K={kk,kk+1}, lanes 16-31 K={kk+2,kk+3}
        v2f a = *(const v2f*)(Ar + ka);
        const float* __restrict__ w0 = &WT[ka * WSTRIDE];
        const float* __restrict__ w1 = &WT[(ka + 1) * WSTRIDE];
        v2f b;
        b.x = w0[l15];        b.y = w1[l15];
        c0 = __builtin_amdgcn_wmma_f32_16x16x4_f32(false, a, false, b, (short)0, c0, false, false);
        b.x = w0[l15 + 16];   b.y = w1[l15 + 16];
        c1 = __builtin_amdgcn_wmma_f32_16x16x4_f32(false, a, false, b, (short)0, c1, false, false);
        b.x = w0[l15 + 32];   b.y = w1[l15 + 32];
        c2 = __builtin_amdgcn_wmma_f32_16x16x4_f32(false, a, false, b, (short)0, c2, false, false);
        b.x = w0[l15 + 48];   b.y = w1[l15 + 48];
        c3 = __builtin_amdgcn_wmma_f32_16x16x4_f32(false, a, false, b, (short)0, c3, false, false);
      }
    }
  }

  if (!active) return;
#pragma unroll
  for (int v = 0; v < 8; ++v) {
    int m = v + (hl << 3);  // C/D layout: VGPR v, lanes>=16 hold M=v+8
    float* __restrict__ o = out + (long long)(row0 + m) * FDIM;
    if (accumulate) {
      o[l15]      += c0[v];
      o[l15 + 16] += c1[v];
      o[l15 + 32] += c2[v];
      o[l15 + 48] += c3[v];
    } else {
      float b0 = bias ? bias[l15]      : 0.0f;
      float b1 = bias ? bias[l15 + 16] : 0.0f;
      float b2 = bias ? bias[l15 + 32] : 0.0f;
      float b3 = bias ? bias[l15 + 48] : 0.0f;
      o[l15]      = c0[v] + b0;
      o[l15 + 16] = c1[v] + b1;
      o[l15 + 32] = c2[v] + b2;
      o[l15 + 48] = c3[v] + b3;
    }
  }
}

// ---------------------------------------------------------------- batchnorm
__global__ __launch_bounds__(256) void bn_stats_kernel(const float* __restrict__ h,
                                                       float* __restrict__ stats, int nrows) {
  int col  = threadIdx.x & 63;
  int rgrp = threadIdx.x >> 6;  // 0..3 rows per block iteration
  float s = 0.0f, s2 = 0.0f;
  for (long long row = (long long)blockIdx.x * 4 + rgrp; row < nrows;
       row += (long long)gridDim.x * 4) {
    float v = h[row * FDIM + col];
    s += v; s2 += v * v;
  }
  __shared__ float sh[256], sh2[256];
  sh[threadIdx.x] = s; sh2[threadIdx.x] = s2;
  __syncthreads();
  if (rgrp == 0) {
    s  = sh[col]  + sh[64 + col]  + sh[128 + col]  + sh[192 + col];
    s2 = sh2[col] + sh2[64 + col] + sh2[128 + col] + sh2[192 + col];
    atomicAdd(&stats[col], s);
    atomicAdd(&stats[64 + col], s2);
  }
}

__global__ __launch_bounds__(256) void bn_relu_kernel(float* __restrict__ h,
                                                      const float* __restrict__ stats,
                                                      const float* __restrict__ g,
                                                      const float* __restrict__ be,
                                                      long long n) {
  long long tid = (long long)blockIdx.x * 256 + threadIdx.x;
  if (tid >= n) return;
  int col = (int)(tid & 63);
  const float inv_n = 1.0f / (float)NN;
  float mu  = stats[col] * inv_n;
  float var = stats[64 + col] * inv_n - mu * mu;
  float v = (h[tid] - mu) * rsqrtf(var + 1e-5f) * g[col] + be[col];
  h[tid] = fmaxf(v, 0.0f);
}

// ---------------------------------------------------------------- classifier
__global__ __launch_bounds__(256) void final_kernel(const float* __restrict__ h,
                                                    const float* __restrict__ Wf,
                                                    const float* __restrict__ bf,
                                                    float* __restrict__ out, int nrows) {
  int n = blockIdx.x * 256 + threadIdx.x;
  if (n >= nrows) return;
  const float* __restrict__ hr = h + (long long)n * FDIM;
  float l0 = bf[0], l1 = bf[1];
#pragma unroll
  for (int f = 0; f < FDIM; ++f) {
    float v = hr[f];
    l0 += v * Wf[f * 2];
    l1 += v * Wf[f * 2 + 1];
  }
  float m = fmaxf(l0, l1);
  float lse = m + logf(expf(l0 - m) + expf(l1 - m));
  out[(long long)n * 2]     = l0 - lse;
  out[(long long)n * 2 + 1] = l1 - lse;
}

// ---------------------------------------------------------------- host side
static void run_layer(const float* xin, const float* W, const float* root,
                      const float* bias, const float* g, const float* be,
                      float* hout, float* S, float* cnt, float* stats, int rch,
                      const int* src, const int* dst, const int* et, int E,
                      hipStream_t stream) {
  int tiles = NN / 16;                 // 6250 (exact)
  int gblocks = (tiles + 7) / 8;
  // hout = x @ root + b   (dense WMMA GEMM, sets initial value)
  gemm16_kernel<<<gblocks, 256, 0, stream>>>(xin, 0, 1, root, bias, hout, NN, 0);
  // relation-chunked scatter-mean + batched WMMA GEMM accumulation
  for (int rb = 0; rb < RREL; rb += rch) {
    int rc = RREL - rb < rch ? RREL - rb : rch;
    long long sn = (long long)rc * NN * FDIM;
    zero_kernel<<<(int)((sn + 255) / 256), 256, 0, stream>>>(S, sn);
    long long st = (long long)E * 64;
    scatter_kernel<<<(int)((st + 255) / 256), 256, 0, stream>>>(
        xin, src, dst, et, cnt, S, E, rb, rc);
    gemm16_kernel<<<gblocks, 256, 0, stream>>>(
        S, (long long)NN * FDIM, rc, W + (long long)rb * FDIM * FDIM,
        nullptr, hout, NN, 1);
  }
  // batchnorm + relu (population var, eps=1e-5)
  zero_kernel<<<1, 256, 0, stream>>>(stats, 128);
  bn_stats_kernel<<<512, 256, 0, stream>>>(hout, stats, NN);
  long long tot = (long long)NN * FDIM;
  bn_relu_kernel<<<(int)((tot + 255) / 256), 256, 0, stream>>>(hout, stats, g, be, tot);
}

extern "C" void kernel_launch(void* const* d_in, const int* in_sizes, int n_in,
                              void* d_out, int out_size, void* d_ws, size_t ws_size,
                              hipStream_t stream) {
  const float* x     = (const float*)d_in[0];
  const int*   ei    = (const int*)d_in[1];   // [2, E] flat: src then dst
  const int*   et    = (const int*)d_in[2];   // [E]
  const float* W1    = (const float*)d_in[3];
  const float* root1 = (const float*)d_in[4];
  const float* b1    = (const float*)d_in[5];
  const float* g1    = (const float*)d_in[6];
  const float* be1   = (const float*)d_in[7];
  const float* W2    = (const float*)d_in[8];
  const float* root2 = (const float*)d_in[9];
  const float* b2    = (const float*)d_in[10];
  const float* g2    = (const float*)d_in[11];
  const float* be2   = (const float*)d_in[12];
  const float* Wf    = (const float*)d_in[13];
  const float* bf    = (const float*)d_in[14];
  float* out = (float*)d_out;

  int E = in_sizes[2];
  const int* src = ei;
  const int* dst = ei + E;

  // workspace layout: cnt[R*N] | h1[N*64] | h2[N*64] | stats[128] | S[rch*N*64]
  float* ws    = (float*)d_ws;
  float* cnt   = ws;
  float* h1    = cnt + (long long)RREL * NN;
  float* h2    = h1 + (long long)NN * FDIM;
  float* stats = h2 + (long long)NN * FDIM;
  float* S     = stats + 128;

  long long avail = (long long)(ws_size / sizeof(float));
  long long fixed = (long long)RREL * NN + 2LL * NN * FDIM + 128;
  long long per   = (long long)NN * FDIM;
  int rch = (int)((avail - fixed) / per);
  if (rch > RREL) rch = RREL;
  if (rch < 1)    rch = 1;   // minimum viable chunk

  // per-(relation,dst) edge counts (shared by both layers)
  zero_kernel<<<(int)(((long long)RREL * NN + 255) / 256), 256, 0, stream>>>(
      cnt, (long long)RREL * NN);
  count_kernel<<<(E + 255) / 256, 256, 0, stream>>>(dst, et, cnt, E);

  run_layer(x,  W1, root1, b1, g1, be1, h1, S, cnt, stats, rch, src, dst, et, E, stream);
  run_layer(h1, W2, root2, b2, g2, be2, h2, S, cnt, stats, rch, src, dst, et, E, stream);

  final_kernel<<<(NN + 255) / 256, 256, 0, stream>>>(h2, Wf, bf, out, NN);
  (void)n_in; (void)out_size;
}